// GCE_GNN_Model_65077344469223
// MI455X (gfx1250) — compile-verified
//
#include <hip/hip_runtime.h>
#include <hip/hip_bf16.h>
#include <math.h>

typedef __attribute__((ext_vector_type(2))) float v2f;
typedef __attribute__((ext_vector_type(8))) float v8f;

#define BB   256
#define LL   64
#define DD   100
#define SS   12
#define NOUT 99999
#define LRELU 0.2f
#define NEGBIG -9e15f

// ---------------------------------------------------------------------------
// h = emb[items]   [B,L,D]
__global__ void k_gather_h(const float* __restrict__ emb, const int* __restrict__ items,
                           float* __restrict__ h) {
    int idx = blockIdx.x * blockDim.x + threadIdx.x;
    int total = BB * LL * DD;
    if (idx >= total) return;
    int d  = idx % DD;
    int bl = idx / DD;
    h[idx] = emb[(size_t)items[bl] * DD + d];
}

// ---------------------------------------------------------------------------
// sess[b,d] = sum_l emb[seq_features[b,l],d]*mask[b,l] / sum_l mask[b,l]
__global__ void k_sess(const float* __restrict__ emb, const int* __restrict__ seqf,
                       const float* __restrict__ mask, float* __restrict__ sess) {
    int b = blockIdx.x;
    float msum = 0.f;
    for (int l = 0; l < LL; ++l) msum += mask[b * LL + l];
    for (int d = threadIdx.x; d < DD; d += blockDim.x) {
        float acc = 0.f;
        for (int l = 0; l < LL; ++l)
            acc += emb[(size_t)seqf[b * LL + l] * DD + d] * mask[b * LL + l];
        sess[b * DD + d] = acc / msum;
    }
}

// ---------------------------------------------------------------------------
// Local aggregator: one block (64 threads) per (b,i)
__global__ void k_local(const float* __restrict__ h, const int* __restrict__ adj,
                        const float* __restrict__ a0, const float* __restrict__ a1,
                        const float* __restrict__ a2, const float* __restrict__ a3,
                        float* __restrict__ h_local) {
    int bi = blockIdx.x;
    int b = bi / LL, i = bi % LL;
    __shared__ float hi[DD];
    __shared__ float red[LL];
    __shared__ float w[LL];
    const float* hb = h + (size_t)b * LL * DD;
    for (int d = threadIdx.x; d < DD; d += blockDim.x) hi[d] = hb[i * DD + d];
    __syncthreads();

    int j = threadIdx.x;
    int a = adj[((size_t)b * LL + i) * LL + j];
    float alpha;
    if (a >= 1 && a <= 4) {
        const float* av = (a == 1) ? a0 : (a == 2) ? a1 : (a == 3) ? a2 : a3;
        const float* hj = hb + j * DD;
        float s = 0.f;
        for (int d = 0; d < DD; ++d) s += hi[d] * hj[d] * av[d];
        alpha = (s > 0.f) ? s : LRELU * s;
    } else {
        alpha = NEGBIG;
    }
    red[j] = alpha; __syncthreads();
    for (int off = 32; off > 0; off >>= 1) {
        if (j < off) red[j] = fmaxf(red[j], red[j + off]);
        __syncthreads();
    }
    float mx = red[0]; __syncthreads();
    float e = __expf(alpha - mx);
    red[j] = e; __syncthreads();
    for (int off = 32; off > 0; off >>= 1) {
        if (j < off) red[j] += red[j + off];
        __syncthreads();
    }
    w[j] = e / red[0]; __syncthreads();

    for (int d = threadIdx.x; d < DD; d += blockDim.x) {
        float acc = 0.f;
        for (int jj = 0; jj < LL; ++jj) acc += w[jj] * hb[jj * DD + d];
        h_local[((size_t)b * LL + i) * DD + d] = acc;
    }
}

// ---------------------------------------------------------------------------
// Global aggregator + combine: one block (128 threads) per (b,l); in-place on h_comb
__global__ void k_global(const float* __restrict__ emb, const float* __restrict__ h,
                         const float* __restrict__ sess, const int* __restrict__ items,
                         const int* __restrict__ adj_all, const float* __restrict__ num_w,
                         const float* __restrict__ gw1, const float* __restrict__ gw2,
                         const float* __restrict__ gw3, float* __restrict__ h_comb) {
    int bl = blockIdx.x;
    int b = bl / LL, l = bl % LL;
    __shared__ float s_sess[DD], s_hi[DD], agg[DD];
    __shared__ float nbr[SS][DD];
    __shared__ float att[SS][DD];
    __shared__ float wsm[SS], nw[SS];
    __shared__ int nidx[SS];
    int tid = threadIdx.x;
    int it = items[b * LL + l];
    if (tid < SS) {
        nidx[tid] = adj_all[(size_t)it * SS + tid];
        nw[tid]   = num_w[(size_t)it * SS + tid];
    }
    for (int d = tid; d < DD; d += blockDim.x) {
        s_sess[d] = sess[b * DD + d];
        s_hi[d]   = h[((size_t)b * LL + l) * DD + d];
    }
    __syncthreads();
    for (int x = tid; x < SS * DD; x += blockDim.x) {
        int s = x / DD, d = x % DD;
        nbr[s][d] = emb[(size_t)nidx[s] * DD + d];
    }
    __syncthreads();
    // att = leaky_relu([sess*nbr, nbr_w] @ gw1)
    for (int x = tid; x < SS * DD; x += blockDim.x) {
        int s = x / DD, dc = x % DD;
        float acc = nw[s] * gw1[DD * DD + dc];  // row 100 of gw1 (101x100)
        for (int dp = 0; dp < DD; ++dp)
            acc += s_sess[dp] * nbr[s][dp] * gw1[dp * DD + dc];
        att[s][dc] = (acc > 0.f) ? acc : LRELU * acc;
    }
    __syncthreads();
    if (tid < SS) {
        float acc = 0.f;
        for (int d = 0; d < DD; ++d) acc += att[tid][d] * gw2[d];
        wsm[tid] = acc;
    }
    __syncthreads();
    if (tid == 0) {
        float mx = wsm[0];
        for (int s = 1; s < SS; ++s) mx = fmaxf(mx, wsm[s]);
        float sum = 0.f;
        for (int s = 0; s < SS; ++s) { wsm[s] = __expf(wsm[s] - mx); sum += wsm[s]; }
        for (int s = 0; s < SS; ++s) wsm[s] /= sum;
    }
    __syncthreads();
    for (int d = tid; d < DD; d += blockDim.x) {
        float acc = 0.f;
        for (int s = 0; s < SS; ++s) acc += wsm[s] * nbr[s][d];
        agg[d] = acc;
    }
    __syncthreads();
    // h_global = relu([h, agg] @ gw3); h_comb = h_local + h_global (in place)
    for (int d2 = tid; d2 < DD; d2 += blockDim.x) {
        float acc = 0.f;
        for (int d = 0; d < DD; ++d) acc += s_hi[d] * gw3[d * DD + d2];
        for (int d = 0; d < DD; ++d) acc += agg[d] * gw3[(DD + d) * DD + d2];
        float hg = (acc > 0.f) ? acc : 0.f;
        size_t o = ((size_t)b * LL + l) * DD + d2;
        h_comb[o] = h_comb[o] + hg;   // h_comb currently holds h_local
    }
}

// ---------------------------------------------------------------------------
// seq_hidden[b,l,:] = h_comb[b, alias[b,l], :]
__global__ void k_seqhidden(const float* __restrict__ h_comb, const int* __restrict__ alias,
                            float* __restrict__ seqh) {
    int idx = blockIdx.x * blockDim.x + threadIdx.x;
    int total = BB * LL * DD;
    if (idx >= total) return;
    int d  = idx % DD;
    int bl = idx / DD;
    int b  = bl / LL, l = bl % LL;
    seqh[idx] = h_comb[((size_t)b * LL + alias[b * LL + l]) * DD + d];
}

// ---------------------------------------------------------------------------
// hs[b,d] = sum_l seqh*mask / sum_l mask
__global__ void k_hs(const float* __restrict__ seqh, const float* __restrict__ mask,
                     float* __restrict__ hs) {
    int b = blockIdx.x;
    float msum = 0.f;
    for (int l = 0; l < LL; ++l) msum += mask[b * LL + l];
    for (int d = threadIdx.x; d < DD; d += blockDim.x) {
        float acc = 0.f;
        for (int l = 0; l < LL; ++l)
            acc += seqh[((size_t)b * LL + l) * DD + d] * mask[b * LL + l];
        hs[b * DD + d] = acc / msum;
    }
}

// hsg[b,d] = hs[b,:] @ glu2_w[:,d] + glu2_b[d]
__global__ void k_hsg(const float* __restrict__ hs, const float* __restrict__ glu2_w,
                      const float* __restrict__ glu2_b, float* __restrict__ hsg) {
    int b = blockIdx.x;
    for (int d = threadIdx.x; d < DD; d += blockDim.x) {
        float acc = glu2_b[d];
        for (int dp = 0; dp < DD; ++dp) acc += hs[b * DD + dp] * glu2_w[dp * DD + d];
        hsg[b * DD + d] = acc;
    }
}

// ---------------------------------------------------------------------------
// beta[b,l] = (sigmoid(tanh([pos,seqh]@w1)@glu1_w + hsg[b]) @ w2) * mask
__global__ void k_nh(const float* __restrict__ seqh, const float* __restrict__ pos_table,
                     const float* __restrict__ w1, const float* __restrict__ glu1_w,
                     const float* __restrict__ hsg, const float* __restrict__ w2,
                     const float* __restrict__ mask, float* __restrict__ beta) {
    int bl = blockIdx.x;
    int b = bl / LL, l = bl % LL;
    __shared__ float sh[DD], sp[DD], nh1[DD], nh2[DD];
    __shared__ float partial[128];
    int tid = threadIdx.x;
    for (int d = tid; d < DD; d += blockDim.x) {
        sh[d] = seqh[((size_t)b * LL + l) * DD + d];
        sp[d] = pos_table[l * DD + d];
    }
    __syncthreads();
    for (int d = tid; d < DD; d += blockDim.x) {
        float acc = 0.f;
        for (int dp = 0; dp < DD; ++dp) acc += sp[dp] * w1[dp * DD + d];
        for (int dp = 0; dp < DD; ++dp) acc += sh[dp] * w1[(DD + dp) * DD + d];
        nh1[d] = tanhf(acc);
    }
    __syncthreads();
    for (int d = tid; d < DD; d += blockDim.x) {
        float acc = hsg[b * DD + d];
        for (int dp = 0; dp < DD; ++dp) acc += nh1[dp] * glu1_w[dp * DD + d];
        nh2[d] = 1.f / (1.f + __expf(-acc));
    }
    __syncthreads();
    float p = 0.f;
    for (int d = tid; d < DD; d += blockDim.x) p += nh2[d] * w2[d];
    partial[tid] = p; __syncthreads();
    for (int off = 64; off > 0; off >>= 1) {
        if (tid < off) partial[tid] += partial[tid + off];
        __syncthreads();
    }
    if (tid == 0) beta[b * LL + l] = partial[0] * mask[b * LL + l];
}

// select[b,d] = sum_l beta[b,l]*seqh[b,l,d]
__global__ void k_select(const float* __restrict__ beta, const float* __restrict__ seqh,
                         float* __restrict__ sel) {
    int b = blockIdx.x;
    for (int d = threadIdx.x; d < DD; d += blockDim.x) {
        float acc = 0.f;
        for (int l = 0; l < LL; ++l)
            acc += beta[b * LL + l] * seqh[((size_t)b * LL + l) * DD + d];
        sel[b * DD + d] = acc;
    }
}

// ---------------------------------------------------------------------------
// Score GEMM via V_WMMA_F32_16X16X4_F32:  score[m,n] = sum_k sel[m,k]*emb[1+n,k]
// One wave computes a 16x16 tile; A (select) frags live in VGPRs for the whole
// N-sweep (k-loop FULLY unrolled so afrag indices are compile-time constants —
// avoids the scratch spill seen when the loop was only partially unrolled).
#define TILES_PER_WAVE 10
#define NTILES 6250   // ceil(99999/16)
#define KSTEPS 25     // 100 / 4

__global__ void k_score_wmma(const float* __restrict__ sel, const float* __restrict__ emb,
                             float* __restrict__ out) {
    const int lane  = threadIdx.x & 31;
    const int wave  = threadIdx.x >> 5;
    const int wavesPerBlock = blockDim.x >> 5;
    const int mTile = blockIdx.y;                        // 0..15
    const int nTileBase = (blockIdx.x * wavesPerBlock + wave) * TILES_PER_WAVE;

    const int m     = lane & 15;
    const int khalf = lane >> 4;

    // Preload A fragments: A[m][k] for this 16-row strip of `select`
    v2f afrag[KSTEPS];
    const float* arow = sel + (size_t)(mTile * 16 + m) * DD + khalf * 2;
#pragma unroll
    for (int kk = 0; kk < KSTEPS; ++kk) {
        afrag[kk].x = arow[kk * 4];
        afrag[kk].y = arow[kk * 4 + 1];
    }

    for (int t = 0; t < TILES_PER_WAVE; ++t) {
        int nTile = nTileBase + t;
        if (nTile >= NTILES) return;
        int n0 = nTile * 16;
        int n  = n0 + (lane & 15);
        int nc = (n < NOUT) ? n : (NOUT - 1);            // clamp last partial tile
        const float* brow = emb + (size_t)(1 + nc) * DD + khalf * 2;
        __builtin_prefetch(brow + DD * 16, 0, 0);        // pull ahead in L2/WGP$

        v8f acc = {0.f, 0.f, 0.f, 0.f, 0.f, 0.f, 0.f, 0.f};
#pragma unroll
        for (int kk = 0; kk < KSTEPS; ++kk) {
            v2f bfrag;
            bfrag.x = brow[kk * 4];
            bfrag.y = brow[kk * 4 + 1];
            acc = __builtin_amdgcn_wmma_f32_16x16x4_f32(
                false, afrag[kk], false, bfrag, (short)0, acc, false, false);
        }
        // C/D layout: VGPR r -> row r (lanes 0-15) / row r+8 (lanes 16-31), col = lane%16
        int col = lane & 15;
        int nn  = n0 + col;
        if (nn < NOUT) {
            float* orow = out + (size_t)(mTile * 16 + 8 * khalf) * NOUT + nn;
#pragma unroll
            for (int r = 0; r < 8; ++r) {
                orow[(size_t)r * NOUT] = acc[r];
            }
        }
    }
}

// ---------------------------------------------------------------------------
// Online-softmax row statistics over 99999 columns
__global__ void k_rowstats(const float* __restrict__ out, float* __restrict__ rowmax,
                           float* __restrict__ rowsum) {
    int b = blockIdx.x;
    const float* row = out + (size_t)b * NOUT;
    float m = -INFINITY, s = 0.f;
    for (int i = threadIdx.x; i < NOUT; i += blockDim.x) {
        float v = row[i];
        if (v > m) { s = s * __expf(m - v); m = v; }
        s += __expf(v - m);
    }
    __shared__ float sm[256], ss[256];
    sm[threadIdx.x] = m; ss[threadIdx.x] = s; __syncthreads();
    for (int off = 128; off > 0; off >>= 1) {
        if (threadIdx.x < off) {
            float m1 = sm[threadIdx.x], s1 = ss[threadIdx.x];
            float m2 = sm[threadIdx.x + off], s2 = ss[threadIdx.x + off];
            float mm = fmaxf(m1, m2);
            sm[threadIdx.x] = mm;
            ss[threadIdx.x] = s1 * __expf(m1 - mm) + s2 * __expf(m2 - mm);
        }
        __syncthreads();
    }
    if (threadIdx.x == 0) { rowmax[b] = sm[0]; rowsum[b] = ss[0]; }
}

__global__ void k_softmax_norm(float* __restrict__ out, const float* __restrict__ rowmax,
                               const float* __restrict__ rowsum) {
    size_t idx = (size_t)blockIdx.x * blockDim.x + threadIdx.x;
    size_t total = (size_t)BB * NOUT;
    if (idx >= total) return;
    int b = (int)(idx / NOUT);
    out[idx] = __expf(out[idx] - rowmax[b]) / rowsum[b];
}

// ---------------------------------------------------------------------------
extern "C" void kernel_launch(void* const* d_in, const int* in_sizes, int n_in,
                              void* d_out, int out_size, void* d_ws, size_t ws_size,
                              hipStream_t stream) {
    const float* emb      = (const float*)d_in[0];
    const float* pos_tab  = (const float*)d_in[1];
    const float* w1       = (const float*)d_in[2];
    const float* w2       = (const float*)d_in[3];
    const float* glu1_w   = (const float*)d_in[4];
    const float* glu2_w   = (const float*)d_in[5];
    const float* glu2_b   = (const float*)d_in[6];
    const float* a0       = (const float*)d_in[7];
    const float* a1       = (const float*)d_in[8];
    const float* a2       = (const float*)d_in[9];
    const float* a3       = (const float*)d_in[10];
    const float* gw1      = (const float*)d_in[11];
    const float* gw2      = (const float*)d_in[12];
    const float* gw3      = (const float*)d_in[13];
    const float* num_w    = (const float*)d_in[14];
    const float* mask     = (const float*)d_in[15];
    const int*   alias    = (const int*)d_in[16];
    const int*   adj      = (const int*)d_in[17];
    const int*   items    = (const int*)d_in[18];
    const int*   seqf     = (const int*)d_in[19];
    const int*   adj_all  = (const int*)d_in[20];
    float* out = (float*)d_out;

    // workspace carve-out (floats)
    float* ws = (float*)d_ws;
    const size_t BLD = (size_t)BB * LL * DD;
    float* h      = ws;            ws += BLD;
    float* h_comb = ws;            ws += BLD;   // h_local, then h_local + h_global
    float* seqh   = ws;            ws += BLD;
    float* sess   = ws;            ws += (size_t)BB * DD;
    float* hs     = ws;            ws += (size_t)BB * DD;
    float* hsg    = ws;            ws += (size_t)BB * DD;
    float* sel    = ws;            ws += (size_t)BB * DD;
    float* beta   = ws;            ws += (size_t)BB * LL;
    float* rowmax = ws;            ws += BB;
    float* rowsum = ws;            ws += BB;

    const int elems = BB * LL * DD;

    k_gather_h<<<(elems + 255) / 256, 256, 0, stream>>>(emb, items, h);
    k_sess<<<BB, 128, 0, stream>>>(emb, seqf, mask, sess);
    k_local<<<BB * LL, 64, 0, stream>>>(h, adj, a0, a1, a2, a3, h_comb);
    k_global<<<BB * LL, 128, 0, stream>>>(emb, h, sess, items, adj_all, num_w,
                                          gw1, gw2, gw3, h_comb);
    k_seqhidden<<<(elems + 255) / 256, 256, 0, stream>>>(h_comb, alias, seqh);
    k_hs<<<BB, 128, 0, stream>>>(seqh, mask, hs);
    k_hsg<<<BB, 128, 0, stream>>>(hs, glu2_w, glu2_b, hsg);
    k_nh<<<BB * LL, 128, 0, stream>>>(seqh, pos_tab, w1, glu1_w, hsg, w2, mask, beta);
    k_select<<<BB, 128, 0, stream>>>(beta, seqh, sel);

    // Score GEMM: grid.y = 16 M-tiles; each wave covers TILES_PER_WAVE N-tiles
    const int wavesNeeded  = (NTILES + TILES_PER_WAVE - 1) / TILES_PER_WAVE; // 625
    const int wavesPerBlk  = 4;                                              // 128 thr
    dim3 sgrid((wavesNeeded + wavesPerBlk - 1) / wavesPerBlk, 16);
    k_score_wmma<<<sgrid, 128, 0, stream>>>(sel, emb, out);

    k_rowstats<<<BB, 256, 0, stream>>>(out, rowmax, rowsum);
    size_t total = (size_t)BB * NOUT;
    k_softmax_norm<<<(unsigned)((total + 255) / 256), 256, 0, stream>>>(out, rowmax, rowsum);
}